// MSDeformAttn_14602888806446
// MI455X (gfx1250) — compile-verified
//
#include <hip/hip_runtime.h>

// ---------------- problem constants ----------------
#define LQ      19700          // sum of H*W over levels
#define BATCH   2
#define MROWS   (BATCH * LQ)   // 39400
#define CDIM    256
#define NHEADS  8
#define HDIM    32
#define NLEV    3
#define NPTS    4
// per-(b,q): 8 heads * 12 points * (x,y,w) = 288 floats
#define TRIP_ROW 288

typedef __attribute__((ext_vector_type(2))) float v2f;
typedef __attribute__((ext_vector_type(8))) float v8f;

static __device__ __forceinline__ int imin(int a, int b) { return a < b ? a : b; }

// ------------------------------------------------------------------
// Weight transpose: Bt[n*K + k] = B[k*N + n].  Tiny (<=64K elems).
// ------------------------------------------------------------------
__global__ __launch_bounds__(256) void transpose_kernel(
    const float* __restrict__ B, float* __restrict__ Bt, int K, int N)
{
    const int t = blockIdx.x * blockDim.x + threadIdx.x;
    if (t >= K * N) return;
    const int k = t / N;
    const int n = t % N;
    Bt[(size_t)n * K + k] = B[t];
}

// ------------------------------------------------------------------
// Generic fp32 WMMA GEMM with pre-transposed B:
//   out = A[M,K] @ Bt[N,K]^T + bias[N]
// Block = 256 threads = 8 waves arranged 4(M) x 2(N); wave tile 32x64
// (2x4 of 16x16 V_WMMA_F32_16X16X4_F32).  Block tile 128x128.
// Per K-step: 6 x global_load_b64 (imm offsets) feed 8 x v_wmma.
// mode 0: out[row*N+col] = val
// mode 1: scatter W_off result   (N=192) into triples[row*288 + ...] x/y slots
// mode 2: scatter W_attn logits  (N= 96) into triples[row*288 + ...] w slots
// ------------------------------------------------------------------
__global__ __launch_bounds__(256) void gemm_f32_wmma(
    const float* __restrict__ A, const float* __restrict__ Bt,
    const float* __restrict__ bias, float* __restrict__ out,
    int M, int N, int K, int mode)
{
    const int wave = threadIdx.x >> 5;
    const int lane = threadIdx.x & 31;
    const int wm = wave >> 1;            // 0..3
    const int wn = wave & 1;             // 0..1
    const int m0 = blockIdx.x * 128 + wm * 32;
    const int n0 = blockIdx.y * 128 + wn * 64;
    const int lrow = lane & 15;
    const int kq   = (lane >> 4) << 1;   // 0 or 2 : which K pair this lane holds

    const int rA0 = imin(m0 + lrow,      M - 1);
    const int rA1 = imin(m0 + 16 + lrow, M - 1);
    const float* Ar0 = A + (size_t)rA0 * K + kq;
    const float* Ar1 = A + (size_t)rA1 * K + kq;

    const float* Br[4];
    int cB[4];
#pragma unroll
    for (int ni = 0; ni < 4; ++ni) {
        cB[ni] = imin(n0 + ni * 16 + lrow, N - 1);
        Br[ni] = Bt + (size_t)cB[ni] * K + kq;
    }

    v8f acc[2][4];
#pragma unroll
    for (int mi = 0; mi < 2; ++mi)
#pragma unroll
        for (int ni = 0; ni < 4; ++ni)
            acc[mi][ni] = (v8f){};

#pragma unroll 8
    for (int k = 0; k < K; k += 4) {
        v2f a0 = *(const v2f*)(Ar0 + k);
        v2f a1 = *(const v2f*)(Ar1 + k);
        v2f b[4];
#pragma unroll
        for (int ni = 0; ni < 4; ++ni)
            b[ni] = *(const v2f*)(Br[ni] + k);
#pragma unroll
        for (int ni = 0; ni < 4; ++ni) {
            acc[0][ni] = __builtin_amdgcn_wmma_f32_16x16x4_f32(false, a0, false, b[ni], (short)0, acc[0][ni], false, false);
            acc[1][ni] = __builtin_amdgcn_wmma_f32_16x16x4_f32(false, a1, false, b[ni], (short)0, acc[1][ni], false, false);
        }
    }

    float bv[4];
#pragma unroll
    for (int ni = 0; ni < 4; ++ni) bv[ni] = bias[cB[ni]];

    const int colBase = lane & 15;
    const int rowOff  = (lane >> 4) * 8;

    for (int mi = 0; mi < 2; ++mi) {
        for (int ni = 0; ni < 4; ++ni) {
            const int col = n0 + ni * 16 + colBase;
            if (col >= N) continue;
            for (int v = 0; v < 8; ++v) {
                const int row = m0 + mi * 16 + rowOff + v;
                if (row >= M) continue;
                const float val = acc[mi][ni][v] + bv[ni];
                if (mode == 0) {
                    out[(size_t)row * N + col] = val;
                } else if (mode == 1) {
                    // col = h*24 + l*8 + p*2 + c   (h,L,P,2 layout)
                    const int h = col / 24, r = col % 24;
                    const int l = r >> 3, pp = (r >> 1) & 3, c = r & 1;
                    out[(size_t)row * TRIP_ROW + (size_t)(h * 12 + l * 4 + pp) * 3 + c] = val;
                } else {
                    // col = h*12 + (l*4+p)
                    const int h = col / 12, r = col % 12;
                    out[(size_t)row * TRIP_ROW + (size_t)(h * 12 + r) * 3 + 2] = val;
                }
            }
        }
    }
}

// ------------------------------------------------------------------
// Per (b,q,h): softmax over 12 logits; convert offsets to absolute
// pixel coords  px = ref_x*W + off_x - 0.5 , py = ref_y*H + off_y - 0.5
// In-place on the triple buffer.
// ------------------------------------------------------------------
__global__ __launch_bounds__(256) void transform_kernel(
    const float* __restrict__ refp, float* __restrict__ triples, int total)
{
    const int t = blockIdx.x * blockDim.x + threadIdx.x;
    if (t >= total) return;                 // total = MROWS * NHEADS
    const int bq = t >> 3;                  // b*Lq + q
    float* row = triples + (size_t)t * 36;
    const float* rp = refp + (size_t)bq * (NLEV * 2);

    float vx[12], vy[12], vw[12];
#pragma unroll
    for (int i = 0; i < 12; ++i) {
        vx[i] = row[i * 3 + 0];
        vy[i] = row[i * 3 + 1];
        vw[i] = row[i * 3 + 2];
    }
    float mx = -1e30f;
#pragma unroll
    for (int i = 0; i < 12; ++i) mx = fmaxf(mx, vw[i]);
    float s = 0.f;
#pragma unroll
    for (int i = 0; i < 12; ++i) { vw[i] = __expf(vw[i] - mx); s += vw[i]; }
    const float inv = 1.0f / s;

    const float Wl[3] = {150.f, 75.f, 38.f};
    const float Hl[3] = {100.f, 50.f, 25.f};
#pragma unroll
    for (int i = 0; i < 12; ++i) {
        const int l = i >> 2;
        row[i * 3 + 0] = rp[l * 2 + 0] * Wl[l] + vx[i] - 0.5f;
        row[i * 3 + 1] = rp[l * 2 + 1] * Hl[l] + vy[i] - 0.5f;
        row[i * 3 + 2] = vw[i] * inv;
    }
}

// ------------------------------------------------------------------
// One wave per (b,q,h); lane = head dim (0..31).
// 12 points x 4 bilinear corners, coalesced 32-float vector gathers.
// ------------------------------------------------------------------
__global__ __launch_bounds__(256) void sample_kernel(
    const float* __restrict__ value, const float* __restrict__ triples,
    float* __restrict__ mid, int total)
{
    const int wave = threadIdx.x >> 5;
    const int lane = threadIdx.x & 31;
    const int gid  = blockIdx.x * 8 + wave;
    if (gid >= total) return;
    const int h  = gid & 7;
    const int bq = gid >> 3;
    const int b  = bq / LQ;

    const float* row = triples + (size_t)gid * 36;
    float px = 0.f, py = 0.f, pw = 0.f;
    if (lane < 12) {
        px = row[lane * 3 + 0];
        py = row[lane * 3 + 1];
        pw = row[lane * 3 + 2];
    }

    const int Wl[3] = {150, 75, 38};
    const int Hl[3] = {100, 50, 25};
    const int St[3] = {0, 15000, 18750};

    // value layout: [B, Lv, 8, 32]
    const float* vb = value + (size_t)b * LQ * CDIM + h * HDIM + lane;
    float acc = 0.f;

#pragma unroll
    for (int j = 0; j < 12; ++j) {
        const float jx = __shfl(px, j, 32);
        const float jy = __shfl(py, j, 32);
        const float jw = __shfl(pw, j, 32);
        const int l = j >> 2;
        const int Wc = Wl[l], Hc = Hl[l], S = St[l];

        const float xf = floorf(jx), yf = floorf(jy);
        const int x0 = (int)xf, y0 = (int)yf;
        const int x1 = x0 + 1,  y1 = y0 + 1;
        const float fx = jx - xf, fy = jy - yf;
        const float w00 = (1.f - fx) * (1.f - fy);
        const float w10 = fx * (1.f - fy);
        const float w01 = (1.f - fx) * fy;
        const float w11 = fx * fy;

        const bool vx0 = (x0 >= 0) & (x0 < Wc);
        const bool vx1 = (x1 >= 0) & (x1 < Wc);
        const bool vy0 = (y0 >= 0) & (y0 < Hc);
        const bool vy1 = (y1 >= 0) & (y1 < Hc);

        const float v00 = (vx0 & vy0) ? vb[(size_t)(S + y0 * Wc + x0) * CDIM] : 0.f;
        const float v10 = (vx1 & vy0) ? vb[(size_t)(S + y0 * Wc + x1) * CDIM] : 0.f;
        const float v01 = (vx0 & vy1) ? vb[(size_t)(S + y1 * Wc + x0) * CDIM] : 0.f;
        const float v11 = (vx1 & vy1) ? vb[(size_t)(S + y1 * Wc + x1) * CDIM] : 0.f;

        acc += jw * (w00 * v00 + w10 * v10 + w01 * v01 + w11 * v11);
    }
    // mid layout: [B*Lq, 256]
    mid[(size_t)bq * CDIM + h * HDIM + lane] = acc;
}

// ------------------------------------------------------------------
extern "C" void kernel_launch(void* const* d_in, const int* in_sizes, int n_in,
                              void* d_out, int out_size, void* d_ws, size_t ws_size,
                              hipStream_t stream) {
    (void)in_sizes; (void)n_in; (void)out_size; (void)ws_size;
    const float* query  = (const float*)d_in[0];
    const float* refp   = (const float*)d_in[1];
    const float* inflat = (const float*)d_in[2];
    const float* W_val  = (const float*)d_in[3];
    const float* b_val  = (const float*)d_in[4];
    const float* W_off  = (const float*)d_in[5];
    const float* b_off  = (const float*)d_in[6];
    const float* W_attn = (const float*)d_in[7];
    const float* b_attn = (const float*)d_in[8];
    const float* W_out  = (const float*)d_in[9];
    const float* b_out  = (const float*)d_in[10];
    float* out = (float*)d_out;

    float* ws      = (float*)d_ws;
    float* value   = ws;                                  // [39400, 256]
    float* triples = value + (size_t)MROWS * CDIM;        // [39400, 288] (x,y,w)
    float* mid     = triples + (size_t)MROWS * TRIP_ROW;  // [39400, 256]
    float* wt_val  = mid + (size_t)MROWS * CDIM;          // [256, 256] transposed
    float* wt_off  = wt_val  + 256 * 256;                 // [192, 256] transposed
    float* wt_attn = wt_off  + 192 * 256;                 // [ 96, 256] transposed
    float* wt_out  = wt_attn +  96 * 256;                 // [256, 256] transposed

    const dim3 blk(256);
    const int M = MROWS;

    // 0) transpose all weight matrices into [N, K] layout
    transpose_kernel<<<(256 * 256 + 255) / 256, blk, 0, stream>>>(W_val,  wt_val,  256, 256);
    transpose_kernel<<<(256 * 192 + 255) / 256, blk, 0, stream>>>(W_off,  wt_off,  256, 192);
    transpose_kernel<<<(256 *  96 + 255) / 256, blk, 0, stream>>>(W_attn, wt_attn, 256,  96);
    transpose_kernel<<<(256 * 256 + 255) / 256, blk, 0, stream>>>(W_out,  wt_out,  256, 256);

    // 1) value = input_flatten @ W_val + b_val
    gemm_f32_wmma<<<dim3((M + 127) / 128, (256 + 127) / 128), blk, 0, stream>>>(
        inflat, wt_val, b_val, value, M, 256, 256, 0);
    // 2) offsets -> (x,y) slots of triples
    gemm_f32_wmma<<<dim3((M + 127) / 128, (192 + 127) / 128), blk, 0, stream>>>(
        query, wt_off, b_off, triples, M, 192, 256, 1);
    // 3) attn logits -> w slots of triples
    gemm_f32_wmma<<<dim3((M + 127) / 128, (96 + 127) / 128), blk, 0, stream>>>(
        query, wt_attn, b_attn, triples, M, 96, 256, 2);
    // 4) softmax + absolute sampling coords (in place)
    const int total = M * NHEADS;
    transform_kernel<<<(total + 255) / 256, blk, 0, stream>>>(refp, triples, total);
    // 5) bilinear deformable sampling -> mid
    sample_kernel<<<(total + 7) / 8, blk, 0, stream>>>(value, triples, mid, total);
    // 6) out = mid @ W_out + b_out
    gemm_f32_wmma<<<dim3((M + 127) / 128, (256 + 127) / 128), blk, 0, stream>>>(
        mid, wt_out, b_out, out, M, 256, 256, 0);
}